// AttentionHead_18210661335292
// MI455X (gfx1250) — compile-verified
//
#include <hip/hip_runtime.h>

typedef __attribute__((ext_vector_type(2))) float v2f;
typedef __attribute__((ext_vector_type(8))) float v8f;

#define D_MODEL 1024
#define SEQ     2048
#define HS      64
#define BATCH   4

// D = A(16x4,f32) * B(4x16,f32) + C(16x16,f32)  -- V_WMMA_F32_16X16X4_F32
static __device__ __forceinline__ v8f wmma4(v2f a, v2f b, v8f c) {
  return __builtin_amdgcn_wmma_f32_16x16x4_f32(false, a, false, b, (short)0, c,
                                               false, false);
}

static __device__ __forceinline__ float rowmax16(float v) {
  v = fmaxf(v, __shfl_xor(v, 1, 16));
  v = fmaxf(v, __shfl_xor(v, 2, 16));
  v = fmaxf(v, __shfl_xor(v, 4, 16));
  v = fmaxf(v, __shfl_xor(v, 8, 16));
  return v;
}
static __device__ __forceinline__ float rowsum16(float v) {
  v += __shfl_xor(v, 1, 16);
  v += __shfl_xor(v, 2, 16);
  v += __shfl_xor(v, 4, 16);
  v += __shfl_xor(v, 8, 16);
  return v;
}

// ---------------------------------------------------------------------------
// Kernel 0: transpose weights into packed wT[3][64][1024] (w[k][c]->wT[m][c][k])
// so every WMMA B-fragment k/k+1 pair is one contiguous b64 load.
// ---------------------------------------------------------------------------
__global__ __launch_bounds__(256) void transpose_w_kernel(
    const float* __restrict__ wq, const float* __restrict__ wk,
    const float* __restrict__ wv, float* __restrict__ wT)
{
  const int idx = blockIdx.x * 256 + threadIdx.x;  // 65536 elements
  const int k = idx >> 6;                          // 0..1023
  const int c = idx & 63;                          // 0..63
  const int src = k * HS + c;
  const int dst = c * D_MODEL + k;
  wT[dst]               = wq[src];
  wT[65536 + dst]       = wk[src];
  wT[2 * 65536 + dst]   = wv[src];
}

// ---------------------------------------------------------------------------
// Kernel 1: QKV projection.  One wave = one (16-row tile, matrix) pair:
// 4 accumulator chains (32 VGPRs) so the scheduler can keep several
// iterations of b64 loads in flight.  V is written TRANSPOSED (VT[b][d][n]).
// ---------------------------------------------------------------------------
__global__ __launch_bounds__(256) void qkv_proj_kernel(
    const float* __restrict__ x,  const float* __restrict__ wT,
    const float* __restrict__ bq, const float* __restrict__ bk,
    const float* __restrict__ bv,
    float* __restrict__ Q, float* __restrict__ K, float* __restrict__ VT)
{
  const int wave = threadIdx.x >> 5;               // 0..7
  const int lane = threadIdx.x & 31;
  const int gw   = blockIdx.x * 8 + wave;          // 0..1535
  const int mat  = gw >> 9;                        // 0:Q 1:K 2:V
  const int rt   = gw & 511;                       // row tile
  const int row0 = rt * 16;
  const int m    = lane & 15;                      // A row / B,C col in tile
  const int hi   = lane >> 4;                      // half-wave -> K pair

  v8f acc[4] = {v8f{0}, v8f{0}, v8f{0}, v8f{0}};

  const float* xrow = x + (size_t)(row0 + m) * D_MODEL;
  const float* w0   = wT + (size_t)mat * (HS * D_MODEL) + (size_t)m * D_MODEL;

  for (int k0 = 0; k0 < D_MODEL; k0 += 4) {
    const int ko = k0 + 2 * hi;
    v2f a = *(const v2f*)(xrow + ko);              // A[m][ko..ko+1]
#pragma unroll
    for (int t = 0; t < 4; ++t) {
      v2f bf = *(const v2f*)(w0 + t * 16 * D_MODEL + ko);
      acc[t] = wmma4(a, bf, acc[t]);
    }
  }

  // Epilogue: C/D layout -> lane holds col n=m, VGPR j holds row j + 8*hi.
  if (mat < 2) {
    float* dst = (mat == 0) ? Q : K;
    const float* bias = (mat == 0) ? bq : bk;
#pragma unroll
    for (int t = 0; t < 4; ++t) {
      const int col = t * 16 + m;
      const float bi = bias[col];
#pragma unroll
      for (int j = 0; j < 8; ++j)
        dst[(size_t)(row0 + j + 8 * hi) * HS + col] = acc[t][j] + bi;
    }
  } else {
    const int bb   = row0 >> 11;                   // batch
    const int rloc = row0 & (SEQ - 1);             // row within batch
    float* vdst = VT + (size_t)bb * HS * SEQ + rloc + 8 * hi;
#pragma unroll
    for (int t = 0; t < 4; ++t) {
      const int col = t * 16 + m;
      const float bi = bv[col];
#pragma unroll
      for (int j = 0; j < 8; ++j)
        vdst[(size_t)col * SEQ + j] = acc[t][j] + bi;   // VT[bb][col][rloc+r]
    }
  }
}

// ---------------------------------------------------------------------------
// Kernel 2: fused flash attention.  One wave per 16-query tile, streaming
// over 128 key tiles with online softmax; scores never leave registers.
// P-tile transpose goes through a private per-wave LDS slice: DS ops from a
// wave execute in order, so only a compiler scheduling fence is needed.
// ---------------------------------------------------------------------------
__global__ __launch_bounds__(128) void flash_attn_kernel(
    const float* __restrict__ Q, const float* __restrict__ K,
    const float* __restrict__ VT, float* __restrict__ out)
{
  __shared__ float p_lds[4][16 * 17];              // per-wave P-tile transpose

  const int wave  = threadIdx.x >> 5;
  const int lane  = threadIdx.x & 31;
  const int gtile = blockIdx.x * 4 + wave;         // 0..511
  const int b     = gtile >> 7;                    // 128 q-tiles per batch
  const int q0    = (gtile & 127) * 16;
  const int m     = lane & 15;
  const int hi    = lane >> 4;
  const size_t base  = (size_t)b * SEQ * HS;       // Q/K/out base
  const size_t vbase = (size_t)b * HS * SEQ;       // VT base

  // Q A-fragments for all 16 K-steps (d = 0..63 step 4), loaded once.
  v2f qf[16];
  const float* qrow = Q + base + (size_t)(q0 + m) * HS;
#pragma unroll
  for (int d = 0; d < 16; ++d)
    qf[d] = *(const v2f*)(qrow + d * 4 + 2 * hi);

  v8f o[4] = {v8f{0}, v8f{0}, v8f{0}, v8f{0}};     // O accum, 4 col tiles
  float rm[8], rl[8];                               // per-row running max/sum
#pragma unroll
  for (int j = 0; j < 8; ++j) { rm[j] = -__builtin_inff(); rl[j] = 0.0f; }

  float* plds = &p_lds[wave][0];
  const float* vcol = VT + vbase + (size_t)m * SEQ; // per-lane V^T row base
  const float scale = 0.125f;                       // 1/sqrt(64)

  for (int kt = 0; kt < SEQ / 16; ++kt) {
    const int k0 = kt * 16;

    // S = Q * K^T : B fragment = contiguous pair from K[k0+m][..]
    v8f s = {};
    const float* krow = K + base + (size_t)(k0 + m) * HS;
#pragma unroll
    for (int d = 0; d < 16; ++d) {
      v2f bf = *(const v2f*)(krow + d * 4 + 2 * hi);
      s = wmma4(qf[d], bf, s);
    }

    // Online softmax update (rows striped: VGPR j <-> row j + 8*hi).
    float corr[8];
#pragma unroll
    for (int j = 0; j < 8; ++j) {
      const float v    = s[j] * scale;
      const float mx   = rowmax16(v);
      const float mnew = fmaxf(rm[j], mx);
      const float p    = __expf(v - mnew);
      const float sum  = rowsum16(p);
      const float c    = __expf(rm[j] - mnew);
      rl[j]  = rl[j] * c + sum;
      rm[j]  = mnew;
      corr[j] = c;
      s[j]   = p;
    }
#pragma unroll
    for (int t = 0; t < 4; ++t)
#pragma unroll
      for (int j = 0; j < 8; ++j)
        o[t][j] *= corr[j];

    // Transpose P (C-layout -> A-layout) through this wave's LDS slice.
    __builtin_amdgcn_wave_barrier();               // keep DS order (no reorder)
#pragma unroll
    for (int j = 0; j < 8; ++j)
      plds[(j + 8 * hi) * 17 + m] = s[j];
    __builtin_amdgcn_wave_barrier();               // stores before loads

    // O += P * V : B fragment = contiguous pair from VT[d][k0+n..]
#pragma unroll
    for (int n0 = 0; n0 < 16; n0 += 4) {
      v2f pa;
      pa.x = plds[m * 17 + n0 + 2 * hi];
      pa.y = plds[m * 17 + n0 + 2 * hi + 1];
      const int nk = k0 + n0 + 2 * hi;
#pragma unroll
      for (int t = 0; t < 4; ++t) {
        v2f bf = *(const v2f*)(vcol + (size_t)t * 16 * SEQ + nk);
        o[t] = wmma4(pa, bf, o[t]);
      }
    }
    __builtin_amdgcn_wave_barrier();               // loads before next stores
  }

  // Epilogue: divide by softmax denominator, store O.
#pragma unroll
  for (int j = 0; j < 8; ++j) {
    const float inv = 1.0f / rl[j];
    const size_t rbase = base + (size_t)(q0 + j + 8 * hi) * HS;
#pragma unroll
    for (int t = 0; t < 4; ++t)
      out[rbase + t * 16 + m] = o[t][j] * inv;
  }
}

// ---------------------------------------------------------------------------
extern "C" void kernel_launch(void* const* d_in, const int* in_sizes, int n_in,
                              void* d_out, int out_size, void* d_ws, size_t ws_size,
                              hipStream_t stream) {
  const float* x  = (const float*)d_in[0];
  const float* wq = (const float*)d_in[1];
  const float* bq = (const float*)d_in[2];
  const float* wk = (const float*)d_in[3];
  const float* bk = (const float*)d_in[4];
  const float* wv = (const float*)d_in[5];
  const float* bv = (const float*)d_in[6];
  float* out = (float*)d_out;

  const size_t qkv_elems = (size_t)BATCH * SEQ * HS;   // 524288 floats each
  float* Q  = (float*)d_ws;
  float* K  = Q + qkv_elems;
  float* VT = K + qkv_elems;            // stored transposed [b][d][n]
  float* wT = VT + qkv_elems;           // packed [3][64][1024]

  // 65536 elements / 256 threads
  transpose_w_kernel<<<256, 256, 0, stream>>>(wq, wk, wv, wT);
  // 1536 (tile, matrix) waves / 8 waves per block
  qkv_proj_kernel<<<192, 256, 0, stream>>>(x, wT, bq, bk, bv, Q, K, VT);
  // 512 query tiles / 4 waves per block
  flash_attn_kernel<<<128, 128, 0, stream>>>(Q, K, VT, out);
}